// APPNP_node_14482629722242
// MI455X (gfx1250) — compile-verified
//
#include <hip/hip_runtime.h>
#include <hip/hip_bf16.h>

#define NUM_ITER 10
#define EMB_DIM 128
#define N_NODES 50000
#define N_EDGES 800000
#define ATOM_VOCAB 119
#define NUM_FEATS 9

typedef __attribute__((ext_vector_type(2))) float v2f;
typedef __attribute__((ext_vector_type(8))) float v8f;

#if defined(__has_builtin)
#if __has_builtin(__builtin_amdgcn_global_load_async_to_lds_b128) && \
    __has_builtin(__builtin_amdgcn_s_wait_asynccnt)
#define USE_ASYNC_LDS 1
#endif
#endif
#ifndef USE_ASYNC_LDS
#define USE_ASYNC_LDS 0
#endif

#if USE_ASYNC_LDS
// Builtin signature (from hipcc diagnostic): (v4i as1*, v4i as3*, Imm offset, Imm cpol)
typedef int v4i __attribute__((__vector_size__(16)));
typedef __attribute__((address_space(1))) v4i GV4I;
typedef __attribute__((address_space(3))) v4i LV4I;
#endif

// ---------------- Atom encoder: h[n,d] = sum_f emb[f, x[n,f], d] ----------------
__global__ __launch_bounds__(EMB_DIM) void encode_kernel(
    const int* __restrict__ x, const float* __restrict__ emb,
    float* __restrict__ h)
{
    int n = blockIdx.x;
    int d = threadIdx.x;
    float s = 0.f;
#pragma unroll
    for (int f = 0; f < NUM_FEATS; ++f) {
        int id = x[n * NUM_FEATS + f];   // uniform -> scalar load
        s += emb[(f * ATOM_VOCAB + id) * EMB_DIM + d];
    }
    h[n * EMB_DIM + d] = s;
}

// ---------------- GEMM with fp32 WMMA: C = relu?(A[N,128] @ W[128,128] + b) ----------------
// 1 block = 256 threads = 8 waves; block handles 16 rows; wave w -> column tile w.
// V_WMMA_F32_16X16X4_F32 fragment layout (ISA 7.12.2):
//   A/B: lane l, VGPR v  ->  K = kbase + v + 2*(l/16),  M (or N) = l%16
//   C/D: VGPR j          ->  M = j + 8*(l/16),           N = l%16
template <bool RELU, bool DUAL>
__global__ __launch_bounds__(256) void gemm_wmma_kernel(
    const float* __restrict__ A, const float* __restrict__ W,
    const float* __restrict__ bias, float* __restrict__ C,
    float* __restrict__ C2)
{
    __shared__ float Alds[16 * EMB_DIM];
    const int rowbase = blockIdx.x * 16;
    const int tid = threadIdx.x;

#if USE_ASYNC_LDS
    // Stage the 16x128 A tile (8 KB) via the CDNA5 async-to-LDS datapath:
    // 256 lanes x 2 x b128 = 8 KB, tracked on ASYNCcnt (no VGPR bounce).
    {
        const char* gbase = (const char*)(A + (size_t)rowbase * EMB_DIM);
#pragma unroll
        for (int i = 0; i < 2; ++i) {
            const size_t off = (size_t)(tid + 256 * i) * 16;
            __builtin_amdgcn_global_load_async_to_lds_b128(
                (GV4I*)(gbase + off), (LV4I*)((char*)Alds + off), 0, 0);
        }
        __builtin_amdgcn_s_wait_asynccnt(0);
    }
    __syncthreads();
#else
    for (int i = tid; i < 16 * EMB_DIM; i += 256) {
        Alds[i] = A[rowbase * EMB_DIM + i];
    }
    __syncthreads();
#endif

    const int wave = tid >> 5;        // 0..7 : output column tile
    const int lane = tid & 31;
    const int half = lane >> 4;       // 0 or 1
    const int m    = lane & 15;       // row-in-tile for A, col-in-tile for B
    const int colbase = wave * 16;

    v8f acc = {};
#pragma unroll 4
    for (int k = 0; k < EMB_DIM; k += 4) {
        const int ka = k + 2 * half;
        v2f a, b;
        a.x = Alds[m * EMB_DIM + ka];
        a.y = Alds[m * EMB_DIM + ka + 1];
        b.x = W[ka       * EMB_DIM + colbase + m];
        b.y = W[(ka + 1) * EMB_DIM + colbase + m];
        acc = __builtin_amdgcn_wmma_f32_16x16x4_f32(
            /*neg_a=*/false, a, /*neg_b=*/false, b,
            /*c_mod=*/(short)0, acc, /*reuse_a=*/false, /*reuse_b=*/false);
    }

    const int col = colbase + m;
    const float bv = bias[col];
#pragma unroll
    for (int j = 0; j < 8; ++j) {
        const int row = rowbase + j + 8 * half;
        float v = acc[j] + bv;
        if (RELU) v = fmaxf(v, 0.f);
        C[row * EMB_DIM + col] = v;
        if (DUAL) C2[row * EMB_DIM + col] = v;
    }
}

// ---------------- Degree / dinv / coef ----------------
__global__ __launch_bounds__(256) void deg_init_kernel(float* __restrict__ deg, int n)
{
    int i = blockIdx.x * blockDim.x + threadIdx.x;
    if (i < n) deg[i] = 1.0f;   // self loop
}

__global__ __launch_bounds__(256) void deg_count_kernel(
    const int* __restrict__ dst, float* __restrict__ deg, int e)
{
    int i = blockIdx.x * blockDim.x + threadIdx.x;
    if (i < e) unsafeAtomicAdd(&deg[dst[i]], 1.0f);
}

__global__ __launch_bounds__(256) void dinv_kernel(
    const float* __restrict__ deg, float* __restrict__ dinv, int n)
{
    int i = blockIdx.x * blockDim.x + threadIdx.x;
    if (i < n) {
        float d = deg[i];
        dinv[i] = (d > 0.f) ? rsqrtf(fmaxf(d, 1e-12f)) : 0.f;
    }
}

__global__ __launch_bounds__(256) void coef_kernel(
    const int* __restrict__ src, const int* __restrict__ dst,
    const float* __restrict__ dinv, float* __restrict__ coef, int e)
{
    int i = blockIdx.x * blockDim.x + threadIdx.x;
    if (i < e) coef[i] = dinv[src[i]] * dinv[dst[i]];
}

// ---------------- Propagation scatter: one wave32 per edge ----------------
// lane loads float4 of h[src] (global_load_b128) and does 4 global_atomic_add_f32.
__global__ __launch_bounds__(256) void scatter_kernel(
    const float* __restrict__ h, const float* __restrict__ coef,
    const int* __restrict__ src, const int* __restrict__ dst,
    float* __restrict__ agg, int e)
{
    int wid = blockIdx.x * (blockDim.x >> 5) + (threadIdx.x >> 5);
    if (wid >= e) return;
    const int lane = threadIdx.x & 31;
    const int s = src[wid];
    const int d = dst[wid];
    const float c = coef[wid];
    const float4 v = ((const float4*)(h + (size_t)s * EMB_DIM))[lane];
    float* out = agg + (size_t)d * EMB_DIM + lane * 4;
    unsafeAtomicAdd(out + 0, c * v.x);
    unsafeAtomicAdd(out + 1, c * v.y);
    unsafeAtomicAdd(out + 2, c * v.z);
    unsafeAtomicAdd(out + 3, c * v.w);
}

// ---------------- Combine: h = 0.9*(agg + dinv[n]^2*h) + 0.1*h0 ----------------
// Also re-zeroes agg in the same pass (saves a 25.6 MB memset per iteration).
__global__ __launch_bounds__(256) void combine_kernel(
    float* __restrict__ h, float* __restrict__ agg,
    const float* __restrict__ h0, const float* __restrict__ dinv, int total)
{
    int i = blockIdx.x * blockDim.x + threadIdx.x;
    if (i >= total) return;
    const int n = i >> 7;  // EMB_DIM == 128
    const float di = dinv[n];
    const float a = agg[i];
    agg[i] = 0.f;
    h[i] = 0.9f * (a + di * di * h[i]) + 0.1f * h0[i];
}

extern "C" void kernel_launch(void* const* d_in, const int* in_sizes, int n_in,
                              void* d_out, int out_size, void* d_ws, size_t ws_size,
                              hipStream_t stream)
{
    const int*   x    = (const int*)d_in[0];                 // [N, 9] int32
    const int*   ei   = (const int*)d_in[1];                 // [2, E] int32
    const float* emb  = (const float*)d_in[2];               // [9, 119, 128]
    const float* Ws   = (const float*)d_in[3];               // [3, 128, 128]
    const float* bs   = (const float*)d_in[4];               // [3, 128]
    float* h = (float*)d_out;                                // [N, 128] final output

    const int* src = ei;
    const int* dst = ei + N_EDGES;

    // workspace layout (floats)
    const size_t HN = (size_t)N_NODES * EMB_DIM;             // 6.4M
    float* agg  = (float*)d_ws;                              // [N,128] (also GEMM ping buffer)
    float* h0   = agg  + HN;                                 // [N,128]
    float* coef = h0   + HN;                                 // [E]
    float* dinv = coef + N_EDGES;                            // [N]
    float* deg  = dinv + N_NODES;                            // [N]

    // 1) atom encoder -> agg (reused as GEMM scratch before propagation)
    encode_kernel<<<N_NODES, EMB_DIM, 0, stream>>>(x, emb, agg);

    // 2) MLP: ping-pong agg <-> d_out; last layer writes both h0 and h
    const int gemm_blocks = N_NODES / 16;  // 3125
    gemm_wmma_kernel<true,  false><<<gemm_blocks, 256, 0, stream>>>(
        agg, Ws + 0 * EMB_DIM * EMB_DIM, bs + 0 * EMB_DIM, h,  nullptr);
    gemm_wmma_kernel<true,  false><<<gemm_blocks, 256, 0, stream>>>(
        h,   Ws + 1 * EMB_DIM * EMB_DIM, bs + 1 * EMB_DIM, agg, nullptr);
    gemm_wmma_kernel<false, true ><<<gemm_blocks, 256, 0, stream>>>(
        agg, Ws + 2 * EMB_DIM * EMB_DIM, bs + 2 * EMB_DIM, h0, h);

    // 3) gcn_norm: degree (with self loop), dinv, per-edge coef
    deg_init_kernel <<<(N_NODES + 255) / 256, 256, 0, stream>>>(deg, N_NODES);
    deg_count_kernel<<<(N_EDGES + 255) / 256, 256, 0, stream>>>(dst, deg, N_EDGES);
    dinv_kernel     <<<(N_NODES + 255) / 256, 256, 0, stream>>>(deg, dinv, N_NODES);
    coef_kernel     <<<(N_EDGES + 255) / 256, 256, 0, stream>>>(src, dst, dinv, coef, N_EDGES);

    // 4) propagation: zero agg once; combine re-zeroes it each iteration.
    //    Working set ~90 MB -> resident in the 192 MB L2; L2-atomic bound.
    (void)hipMemsetAsync(agg, 0, HN * sizeof(float), stream);
    const int scat_blocks = (N_EDGES + 7) / 8;               // 8 waves/block, 1 edge/wave
    const int comb_blocks = (int)((HN + 255) / 256);
    for (int it = 0; it < NUM_ITER; ++it) {
        scatter_kernel<<<scat_blocks, 256, 0, stream>>>(h, coef, src, dst, agg, N_EDGES);
        combine_kernel<<<comb_blocks, 256, 0, stream>>>(h, agg, h0, dinv, (int)HN);
    }
}